// anchor_54778012893299
// MI455X (gfx1250) — compile-verified
//
#include <hip/hip_runtime.h>
#include <hip/hip_bf16.h>
#include <cstdint>
#include <math.h>

// Problem geometry (fixed by the reference)
#define GDIM 160
#define GG   25600            // GDIM*GDIM
#define NA   3                // anchors
#define CP   86               // channels per anchor (6 + 80 classes)
#define NB   16               // batch
#define STRIDE_F 8.0f         // 1280 / 160

// Kernel tiling
#define TPB 256
#define POS_PER_THREAD 4
#define POS_PER_BLOCK (TPB * POS_PER_THREAD)     // 1024
#define BLOCKS_PER_MAP (GG / POS_PER_BLOCK)      // 25
#define NUM_BLOCKS (NB * NA * BLOCKS_PER_MAP)    // 1200
#define DEPTH 12                                  // async prefetch ring depth (48KB LDS)

// Native clang vector types (required by the amdgcn builtins / nontemporal store)
typedef float v4f __attribute__((ext_vector_type(4)));
typedef int   v4i __attribute__((ext_vector_type(4)));
typedef __attribute__((address_space(1))) v4i* gptr_v4i;   // global int4*
typedef __attribute__((address_space(3))) v4i* lptr_v4i;   // LDS int4*

#if defined(__has_builtin)
#if __has_builtin(__builtin_amdgcn_global_load_async_to_lds_b128)
#define HAVE_ASYNC_B128 1
#endif
#if __has_builtin(__builtin_amdgcn_s_wait_asynccnt)
#define HAVE_WAIT_ASYNC 1
#endif
#if __has_builtin(__builtin_amdgcn_s_wait_dscnt)
#define HAVE_WAIT_DS 1
#endif
#endif

// 16B global -> LDS async copy (CDNA5 async path, tracked by ASYNCcnt)
__device__ __forceinline__ void async_copy_b128(const float* g, v4f* l) {
#if defined(HAVE_ASYNC_B128)
  v4i* gv = (v4i*)g;   // C-style cast: drops const, retypes
  v4i* lv = (v4i*)l;
  __builtin_amdgcn_global_load_async_to_lds_b128(
      (gptr_v4i)gv, (lptr_v4i)lv, /*offset=*/0, /*cpol=*/0);
#else
  unsigned lds_off = (unsigned)(uintptr_t)(__attribute__((address_space(3))) void*)(void*)l;
  asm volatile("global_load_async_to_lds_b128 %0, %1, off"
               :: "v"(lds_off), "v"(g) : "memory");
#endif
}

template <int N>
__device__ __forceinline__ void wait_async_le() {
#if defined(HAVE_WAIT_ASYNC)
  __builtin_amdgcn_s_wait_asynccnt(N);
#else
  asm volatile("s_wait_asynccnt %0" :: "i"(N) : "memory");
#endif
}

__device__ __forceinline__ void wait_ds0() {
#if defined(HAVE_WAIT_DS)
  __builtin_amdgcn_s_wait_dscnt(0);
#else
  asm volatile("s_wait_dscnt 0x0" ::: "memory");
#endif
}

#define COMP(v, j) ((j) == 0 ? (v).x : (j) == 1 ? (v).y : (j) == 2 ? (v).z : (v).w)

__device__ __forceinline__ float fast_sigmoid(float v) {
  return __fdividef(1.0f, 1.0f + __expf(-v));
}

__global__ __launch_bounds__(TPB) void yolo_head_decode(
    const float* __restrict__ x,        // (B, NA*CP, G, G)
    const float* __restrict__ anchors,  // (NA, 2)
    float* __restrict__ out)            // (B, NA*G*G, 7)
{
  __shared__ v4f stage[DEPTH][TPB];  // 48 KB ring: one 4KB channel tile per slot

  const int tid   = threadIdx.x;
  const int blk   = blockIdx.x;
  const int map   = blk / BLOCKS_PER_MAP;   // b*NA + a
  const int chunk = blk % BLOCKS_PER_MAP;
  const int a     = map % NA;
  const int b     = map / NA;
  const int p0    = chunk * POS_PER_BLOCK + tid * POS_PER_THREAD;

  // channel c of this (b,a) map lives at x[(map*CP + c)*GG + p]
  const float* gsrc = x + (size_t)map * CP * GG + p0;

  const float aw = anchors[a * 2 + 0];
  const float ah = anchors[a * 2 + 1];

  // ---- prologue: fill the async ring with channels 0..DEPTH-1 ----
#pragma unroll
  for (int c = 0; c < DEPTH; ++c)
    async_copy_b128(gsrc + (size_t)c * GG, &stage[c][tid]);

  v4f   hdr[6];
  float best[POS_PER_THREAD];
  int   bidx[POS_PER_THREAD];
#pragma unroll
  for (int j = 0; j < POS_PER_THREAD; ++j) { best[j] = -INFINITY; bidx[j] = 0; }

  auto upd = [&](const v4f& v, int cls) {
    if (v.x > best[0]) { best[0] = v.x; bidx[0] = cls; }
    if (v.y > best[1]) { best[1] = v.y; bidx[1] = cls; }
    if (v.z > best[2]) { best[2] = v.z; bidx[2] = cls; }
    if (v.w > best[3]) { best[3] = v.w; bidx[3] = cls; }
  };

  int slot = 0;

  // ---- header channels 0..5 (sigmoid/exp/yaw/conf inputs) ----
#pragma unroll
  for (int c = 0; c < 6; ++c) {
    wait_async_le<DEPTH - 1>();               // oldest channel has landed in LDS
    hdr[c] = stage[slot][tid];
    wait_ds0();                               // LDS read done before slot reuse
    async_copy_b128(gsrc + (size_t)(c + DEPTH) * GG, &stage[slot][tid]);
    if (++slot == DEPTH) slot = 0;
  }

  // ---- steady state: 17 groups of 4 class channels (c = 6..73), batched waits ----
  for (int cg = 6; cg < CP - DEPTH; cg += 4) {
    wait_async_le<DEPTH - 4>();               // 4 oldest tiles have landed
    int s1 = slot + 1; if (s1 >= DEPTH) s1 -= DEPTH;
    int s2 = slot + 2; if (s2 >= DEPTH) s2 -= DEPTH;
    int s3 = slot + 3; if (s3 >= DEPTH) s3 -= DEPTH;
    v4f v0 = stage[slot][tid];
    v4f v1 = stage[s1][tid];
    v4f v2 = stage[s2][tid];
    v4f v3 = stage[s3][tid];
    upd(v0, cg - 6);
    upd(v1, cg - 5);
    upd(v2, cg - 4);
    upd(v3, cg - 3);
    wait_ds0();                               // all 4 LDS reads done before reuse
    async_copy_b128(gsrc + (size_t)(cg + DEPTH + 0) * GG, &stage[slot][tid]);
    async_copy_b128(gsrc + (size_t)(cg + DEPTH + 1) * GG, &stage[s1][tid]);
    async_copy_b128(gsrc + (size_t)(cg + DEPTH + 2) * GG, &stage[s2][tid]);
    async_copy_b128(gsrc + (size_t)(cg + DEPTH + 3) * GG, &stage[s3][tid]);
    slot = s3 + 1; if (slot >= DEPTH) slot = 0;
  }

  // ---- epilogue: last DEPTH channels already in flight; drain once ----
  wait_async_le<0>();
#pragma unroll
  for (int c = CP - DEPTH; c < CP; ++c) {
    v4f v = stage[slot][tid];
    upd(v, c - 6);
    if (++slot == DEPTH) slot = 0;
  }

  // ---- decode + coalesced NT store: 28 contiguous floats per thread ----
  float o[POS_PER_THREAD * 7];
#pragma unroll
  for (int j = 0; j < POS_PER_THREAD; ++j) {
    const int p   = p0 + j;
    const float gx = (float)(p % GDIM);
    const float gy = (float)(p / GDIM);
    const float y0 = COMP(hdr[0], j);
    const float y1 = COMP(hdr[1], j);
    const float y2 = COMP(hdr[2], j);
    const float y3 = COMP(hdr[3], j);
    const float y4 = COMP(hdr[4], j);
    const float y5 = COMP(hdr[5], j);
    o[j * 7 + 0] = floorf((fast_sigmoid(y0) + gx) * STRIDE_F);
    o[j * 7 + 1] = floorf((fast_sigmoid(y1) + gy) * STRIDE_F);
    o[j * 7 + 2] = __expf(y2) * aw;
    o[j * 7 + 3] = __expf(y3) * ah;
    o[j * 7 + 4] = y4;
    o[j * 7 + 5] = fast_sigmoid(y5);
    o[j * 7 + 6] = (float)bidx[j];
  }

  // out[(b*NA*GG + a*GG + p0)*7 ...], 16B-aligned (p0 % 4 == 0 => 112B-aligned)
  v4f* dst = (v4f*)(out + (size_t)((b * NA + a) * GG + p0) * 7);
#pragma unroll
  for (int k = 0; k < 7; ++k) {
    v4f w4 = {o[4 * k + 0], o[4 * k + 1], o[4 * k + 2], o[4 * k + 3]};
    __builtin_nontemporal_store(w4, dst + k);   // streamed output, never re-read
  }
}

extern "C" void kernel_launch(void* const* d_in, const int* in_sizes, int n_in,
                              void* d_out, int out_size, void* d_ws, size_t ws_size,
                              hipStream_t stream) {
  const float* x       = (const float*)d_in[0];
  // d_in[1] = target : unused by the reference output
  const float* anchors = (const float*)d_in[2];
  float* out           = (float*)d_out;
  yolo_head_decode<<<dim3(NUM_BLOCKS), dim3(TPB), 0, stream>>>(x, anchors, out);
}